// ContrastiveAffinityLossWithMemoryV2_18502719111428
// MI455X (gfx1250) — compile-verified
//
#include <hip/hip_runtime.h>
#include <math.h>

#define N_CLASSES 8192
#define BATCH     4096
#define D_FULL    256
#define NUSE      192          // 256 * 0.75
#define EPSF      1e-12f
#define MARGIN    4.0f
#define BG_SIM    0.2f
#define BG_OTHER  0.01f
#define N_PAIRS   8386560.0f   // 4096*4095/2

typedef __attribute__((ext_vector_type(2))) float v2f;
typedef __attribute__((ext_vector_type(8))) float v8f;

#define WMMA_F32(a, b, c) \
    __builtin_amdgcn_wmma_f32_16x16x4_f32(false, (a), false, (b), (short)0, (c), false, false)

// ---------------------------------------------------------------- init accum
__global__ void k_init_accum(float* __restrict__ accum) {
    if (threadIdx.x < 8) accum[threadIdx.x] = 0.0f;
}

// ------------------------------------------------------- row-normalize z
__global__ __launch_bounds__(256) void k_normalize(const float* __restrict__ ypred,
                                                   float* __restrict__ zn) {
    __shared__ float red[256];
    const int row = blockIdx.x;
    const int tid = threadIdx.x;
    float v = 0.0f;
    if (tid < NUSE) v = ypred[(size_t)row * D_FULL + tid];
    red[tid] = v * v;
    __syncthreads();
    for (int s = 128; s > 0; s >>= 1) {
        if (tid < s) red[tid] += red[tid + s];
        __syncthreads();
    }
    const float rn = 1.0f / fmaxf(sqrtf(red[0]), EPSF);
    if (tid < NUSE) zn[(size_t)row * NUSE + tid] = v * rn;
}

// ------------------------------------------------- per-class EMA bank update
// one thread per class; labels staged in LDS (16 KB); per-class chain keeps
// the reference's sequential scan ordering exactly.
__global__ __launch_bounds__(256) void k_bank_update(const int* __restrict__ y,
                                                     const float* __restrict__ zn,
                                                     const int* __restrict__ gstep,
                                                     float* __restrict__ bank,
                                                     float* __restrict__ w) {
    __shared__ int lbl[BATCH];
    const int tid = threadIdx.x;
    for (int i = tid; i < BATCH; i += blockDim.x) lbl[i] = y[i];
    __syncthreads();

    const int c = blockIdx.x * blockDim.x + tid;   // class id, grid == 8192 exactly
    const int gs = gstep[0];
    const float mom = (gs < 5000) ? (0.5f + 0.4f * ((float)gs / 5000.0f)) : 0.9f;
    const float om  = 1.0f - mom;

    float* row = bank + (size_t)c * NUSE;
    bool init = false;
    for (int i = 0; i < BATCH; ++i) {
        if (lbl[i] == c) {
            const float* zi = zn + (size_t)i * NUSE;
            if (!init) {
                for (int k = 0; k < NUSE; ++k) row[k] = zi[k];
                init = true;
            } else {
                float ss = 0.0f;
                for (int k = 0; k < NUSE; ++k) {
                    const float v = mom * row[k] + om * zi[k];
                    ss += v * v;
                }
                const float rn = 1.0f / fmaxf(sqrtf(ss), EPSF);
                for (int k = 0; k < NUSE; ++k) {
                    const float v = mom * row[k] + om * zi[k];
                    row[k] = v * rn;
                }
            }
        }
    }
    if (!init) {
        for (int k = 0; k < NUSE; ++k) row[k] = 0.0f;   // untouched rows are zero
    }
    w[c] = init ? 1.0f : 0.0f;
}

// ----------------------------------------------- sums of w and valid flags
__global__ __launch_bounds__(256) void k_sums(const float* __restrict__ w,
                                              const int* __restrict__ y,
                                              float* __restrict__ accum) {
    __shared__ float r1[256], r2[256];
    const int tid = threadIdx.x;
    float ws = 0.0f, vs = 0.0f;
    for (int i = tid; i < N_CLASSES; i += 256) ws += w[i];
    for (int i = tid; i < BATCH; i += 256) vs += (y[i] >= 0) ? 1.0f : 0.0f;
    r1[tid] = ws; r2[tid] = vs;
    __syncthreads();
    for (int s = 128; s > 0; s >>= 1) {
        if (tid < s) { r1[tid] += r1[tid + s]; r2[tid] += r2[tid + s]; }
        __syncthreads();
    }
    if (tid == 0) { accum[2] = r1[0]; accum[3] = r2[0]; }
}

// ------------------------------------------- epilogue for one 16x16 G tile
__device__ __forceinline__ float pair_tile_loss(const v8f& c, int r0, int c0,
                                                int lane, const int* __restrict__ y,
                                                const float* __restrict__ lookup) {
    const int mr   = lane & 15;
    const int half = lane >> 4;
    const int j  = c0 + mr;                 // N = lane&15
    const int l2 = y[j];
    float sum = 0.0f;
#pragma unroll
    for (int r = 0; r < 8; ++r) {
        const int i = r0 + r + 8 * half;    // M = r + 8*(lane>=16)
        if (i < j) {
            const int   l1 = y[i];
            const float g  = c[r];
            const float d2 = fmaxf(2.0f - 2.0f * g, EPSF);
            const float d  = sqrtf(d2);
            const bool b1 = (l1 == -1), b2 = (l2 == -1);
            float t;
            if (b1 & b2)      t = BG_SIM;
            else if (b1 ^ b2) t = BG_OTHER;
            else              t = lookup[(size_t)l1 * N_CLASSES + l2];
            const float md = fmaxf(MARGIN - d, 0.0f);
            sum += t * d2 + (1.0f - t) * md * md;
        }
    }
    return sum;
}

// ------------------------------------------- epilogue for one 16x16 S tile
__device__ __forceinline__ float mem_tile_loss(const v8f& c, int r0, int c0,
                                               int lane, const int* __restrict__ y,
                                               const float* __restrict__ w,
                                               const float* __restrict__ lookup) {
    const int mr   = lane & 15;
    const int half = lane >> 4;
    const int   j  = c0 + mr;               // class column
    const float wj = w[j];
    float sum = 0.0f;
#pragma unroll
    for (int r = 0; r < 8; ++r) {
        const int i  = r0 + r + 8 * half;   // sample row
        const int yi = y[i];
        if (yi >= 0 && wj != 0.0f) {        // masked terms contribute 0 anyway
            const float s   = c[r];
            const float dm2 = fmaxf(2.0f - 2.0f * s, EPSF);
            const float dm  = sqrtf(dm2);
            const float tm  = lookup[(size_t)yi * N_CLASSES + j]; // contiguous in j
            const float md  = fmaxf(MARGIN - dm, 0.0f);
            sum += wj * (tm * dm2 + (1.0f - tm) * md * md);
        }
    }
    return sum;
}

// -------------------- fused G = zn*zn^T pair loss, 32x32 per wave (2x2 WMMA)
__global__ __launch_bounds__(256) void k_pair_loss(const float* __restrict__ zn,
                                                   const int* __restrict__ y,
                                                   const float* __restrict__ lookup,
                                                   float* __restrict__ accum) {
    const int lane = threadIdx.x & 31;
    const int wave = threadIdx.x >> 5;
    const int tile = blockIdx.x * 8 + wave;     // 128x128 macro-tiles of 32x32
    const int tr = tile >> 7;
    const int tc = tile & 127;
    if (tc < tr) return;                        // wave-uniform: EXEC stays full

    const int row0 = tr << 5, col0 = tc << 5;
    const int mr = lane & 15;
    const int kb = (lane >> 4) << 1;            // 0 or 2
    const float* pa0 = zn + (size_t)(row0 + mr) * NUSE + kb;
    const float* pa1 = pa0 + (size_t)16 * NUSE;
    const float* pb0 = zn + (size_t)(col0 + mr) * NUSE + kb;
    const float* pb1 = pb0 + (size_t)16 * NUSE;

    v8f c00 = {0.f,0.f,0.f,0.f,0.f,0.f,0.f,0.f};
    v8f c01 = c00, c10 = c00, c11 = c00;
#pragma unroll 2
    for (int k0 = 0; k0 < NUSE; k0 += 4) {
        const v2f a0 = *(const v2f*)(pa0 + k0);
        const v2f a1 = *(const v2f*)(pa1 + k0);
        const v2f b0 = *(const v2f*)(pb0 + k0);
        const v2f b1 = *(const v2f*)(pb1 + k0);
        c00 = WMMA_F32(a0, b0, c00);
        c01 = WMMA_F32(a0, b1, c01);
        c10 = WMMA_F32(a1, b0, c10);
        c11 = WMMA_F32(a1, b1, c11);
    }

    float sum = 0.0f;
    sum += pair_tile_loss(c00, row0,      col0,      lane, y, lookup);
    sum += pair_tile_loss(c01, row0,      col0 + 16, lane, y, lookup);
    sum += pair_tile_loss(c10, row0 + 16, col0,      lane, y, lookup);
    sum += pair_tile_loss(c11, row0 + 16, col0 + 16, lane, y, lookup);

    for (int off = 16; off; off >>= 1) sum += __shfl_xor(sum, off, 32);
    if (lane == 0) atomicAdd(&accum[0], sum);
}

// ---------------- fused S = zn*bank^T memory loss, 32x32 per wave (2x2 WMMA)
__global__ __launch_bounds__(256) void k_mem_loss(const float* __restrict__ zn,
                                                  const float* __restrict__ bank,
                                                  const float* __restrict__ w,
                                                  const int* __restrict__ y,
                                                  const float* __restrict__ lookup,
                                                  float* __restrict__ accum) {
    const int lane = threadIdx.x & 31;
    const int wave = threadIdx.x >> 5;
    const int tile = blockIdx.x * 8 + wave;     // 128 x 256 macro-tiles of 32x32
    const int tr = tile >> 8;
    const int tc = tile & 255;

    const int row0 = tr << 5, col0 = tc << 5;
    const int mr = lane & 15;
    const int kb = (lane >> 4) << 1;
    const float* pa0 = zn   + (size_t)(row0 + mr) * NUSE + kb;
    const float* pa1 = pa0 + (size_t)16 * NUSE;
    const float* pb0 = bank + (size_t)(col0 + mr) * NUSE + kb;
    const float* pb1 = pb0 + (size_t)16 * NUSE;

    v8f c00 = {0.f,0.f,0.f,0.f,0.f,0.f,0.f,0.f};
    v8f c01 = c00, c10 = c00, c11 = c00;
#pragma unroll 2
    for (int k0 = 0; k0 < NUSE; k0 += 4) {
        const v2f a0 = *(const v2f*)(pa0 + k0);
        const v2f a1 = *(const v2f*)(pa1 + k0);
        const v2f b0 = *(const v2f*)(pb0 + k0);
        const v2f b1 = *(const v2f*)(pb1 + k0);
        c00 = WMMA_F32(a0, b0, c00);
        c01 = WMMA_F32(a0, b1, c01);
        c10 = WMMA_F32(a1, b0, c10);
        c11 = WMMA_F32(a1, b1, c11);
    }

    float sum = 0.0f;
    sum += mem_tile_loss(c00, row0,      col0,      lane, y, w, lookup);
    sum += mem_tile_loss(c01, row0,      col0 + 16, lane, y, w, lookup);
    sum += mem_tile_loss(c10, row0 + 16, col0,      lane, y, w, lookup);
    sum += mem_tile_loss(c11, row0 + 16, col0 + 16, lane, y, w, lookup);

    for (int off = 16; off; off >>= 1) sum += __shfl_xor(sum, off, 32);
    if (lane == 0) atomicAdd(&accum[1], sum);
}

// ---------------------------------------------------------------- finalize
__global__ void k_finalize(const float* __restrict__ accum,
                           const int* __restrict__ gstep,
                           float* __restrict__ out) {
    const float batch_loss = accum[0] / N_PAIRS;
    const float denom      = fmaxf(accum[3] * accum[2], 1.0f);
    const float mem_loss   = accum[1] / denom;
    const float gs = (float)gstep[0];
    const float progress = fminf(1.0f, (gs - 1000.0f) / 5000.0f);
    const float aw = 0.5f * progress;
    out[0] = (1.0f - aw) * batch_loss + aw * mem_loss;
}

// ---------------------------------------------------------------- launcher
extern "C" void kernel_launch(void* const* d_in, const int* in_sizes, int n_in,
                              void* d_out, int out_size, void* d_ws, size_t ws_size,
                              hipStream_t stream) {
    const int*   y      = (const int*)d_in[0];     // (4096,) int32, -1 = background
    const float* ypred  = (const float*)d_in[1];   // (4096,256) fp32
    const float* lookup = (const float*)d_in[2];   // (8192,8192) fp32
    const int*   gstep  = (const int*)d_in[3];     // scalar int
    float*       out    = (float*)d_out;           // scalar fp32

    // workspace layout (~9.5 MB)
    float* zn    = (float*)d_ws;                        // 4096*192
    float* bank  = zn + (size_t)BATCH * NUSE;           // 8192*192
    float* w     = bank + (size_t)N_CLASSES * NUSE;     // 8192
    float* accum = w + N_CLASSES;                       // 8 scalars

    k_init_accum <<<1, 32, 0, stream>>>(accum);
    k_normalize  <<<BATCH, 256, 0, stream>>>(ypred, zn);
    k_bank_update<<<N_CLASSES / 256, 256, 0, stream>>>(y, zn, gstep, bank, w);
    k_sums       <<<1, 256, 0, stream>>>(w, y, accum);
    k_pair_loss  <<<(128 * 128) / 8, 256, 0, stream>>>(zn, y, lookup, accum);
    k_mem_loss   <<<(128 * 256) / 8, 256, 0, stream>>>(zn, bank, w, y, lookup, accum);
    k_finalize   <<<1, 1, 0, stream>>>(accum, gstep, out);
}